// SetEq4to4_26817775796967
// MI455X (gfx1250) — compile-verified
//
#include <hip/hip_runtime.h>

// ---------------------------------------------------------------------------
// Permutation-equivariant 4->4 set layer, MI455X (gfx1250), wave32.
// All contractions run on V_WMMA_F32_16X16X4_F32 (full fp32 precision).
// ---------------------------------------------------------------------------

typedef __attribute__((ext_vector_type(2))) float v2f;
typedef __attribute__((ext_vector_type(8))) float v8f;

#define P1SZ 4194304   // per 1-axis pool tensor (8*128*4096)
#define P2SZ 262144
#define P3SZ 16384

// Async global->LDS staging (gfx1250) if the toolchain exposes the builtins.
// Probe-confirmed: the builtin exists here with (v4i AS1*, v4i AS3*, imm, imm).
#if __has_builtin(__builtin_amdgcn_global_load_async_to_lds_b128) && \
    __has_builtin(__builtin_amdgcn_s_wait_asynccnt)
#define HAVE_ASYNC_LDS 1
typedef __attribute__((ext_vector_type(4))) int v4i;
typedef __attribute__((address_space(1))) v4i* gv4;   // global memory operand
typedef __attribute__((address_space(3))) v4i* sv4;   // LDS operand
#endif

// ---------------------------------------------------------------------------
// Pack coefs (D,S,69) -> Wp[b][d][s] so WMMA A-fragments read stride-1 in s.
__global__ void pack_w(const float* __restrict__ coefs, float* __restrict__ Wp) {
  int idx = blockIdx.x * 256 + threadIdx.x;          // 69*16384 threads
  int b = idx >> 14;
  int r = idx & 16383;
  int d = r >> 7, s = r & 127;
  Wp[idx] = coefs[(d * 128 + s) * 69 + b];
}

// ---------------------------------------------------------------------------
// Pooling pass A: one read of the (a1,a2,a3) slab for fixed (n,d,a0);
// produces P1_3 (sum over a3) and P1_2 (sum over a2).
__global__ void pool_a(const float* __restrict__ x, float* __restrict__ P1) {
  __shared__ float tile[256 * 17];                   // padded vs bank conflicts
  int wg = blockIdx.x;                               // (n*128+d)*16 + a0
  const float* xb = x + (size_t)wg * 4096;
  int t = threadIdx.x;                               // t = a1*16 + a2
  const float* row = xb + t * 16;                    // 16 consecutive a3
  float sum = 0.f;
#pragma unroll
  for (int i = 0; i < 16; ++i) { float v = row[i]; sum += v; tile[t * 17 + i] = v; }
  P1[(size_t)3 * P1SZ + (size_t)wg * 256 + t] = sum; // P1_3[nd,a0,a1,a2]
  __syncthreads();
  int a1 = t >> 4, a3 = t & 15;
  float s2 = 0.f;
#pragma unroll
  for (int a2 = 0; a2 < 16; ++a2) s2 += tile[(a1 * 16 + a2) * 17 + a3];
  P1[(size_t)2 * P1SZ + (size_t)wg * 256 + t] = s2;  // P1_2[nd,a0,a1,a3]
}

// Pooling pass B: one read of the (a0,a1,a3) slab for fixed (n,d,a2);
// produces P1_1 (sum over a1) and P1_0 (sum over a0).
// Stages the 16KB slab through LDS; uses GLOBAL_LOAD_ASYNC_TO_LDS_B128
// (ASYNCcnt) when the toolchain exposes it.
__global__ void pool_b(const float* __restrict__ x, float* __restrict__ P1) {
#ifdef HAVE_ASYNC_LDS
  constexpr int PITCH = 20;                          // 80B rows: 16B-aligned
#else
  constexpr int PITCH = 17;
#endif
  __shared__ float tile[256 * PITCH];
  int wg = blockIdx.x;                               // (n*128+d)*16 + a2
  int nd = wg >> 4, a2 = wg & 15;
  int t = threadIdx.x;                               // t = a0*16 + a1
  const float* row = x + (size_t)nd * 65536 + (size_t)t * 256 + a2 * 16;
#ifdef HAVE_ASYNC_LDS
#pragma unroll
  for (int q = 0; q < 4; ++q) {
    __builtin_amdgcn_global_load_async_to_lds_b128(
        (gv4)(row + 4 * q), (sv4)(tile + t * PITCH + 4 * q), 0, 0);
  }
  __builtin_amdgcn_s_wait_asynccnt(0);
  __syncthreads();
#else
#pragma unroll
  for (int i = 0; i < 16; ++i) tile[t * PITCH + i] = row[i];
  __syncthreads();
#endif
  int hi = t >> 4, a3 = t & 15;
  float s1 = 0.f;                                    // sum over a1 (hi = a0)
#pragma unroll
  for (int a1 = 0; a1 < 16; ++a1) s1 += tile[(hi * 16 + a1) * PITCH + a3];
  P1[(size_t)1 * P1SZ + (size_t)nd * 4096 + hi * 256 + a2 * 16 + a3] = s1;
  float s0 = 0.f;                                    // sum over a0 (hi = a1)
#pragma unroll
  for (int a0 = 0; a0 < 16; ++a0) s0 += tile[(a0 * 16 + hi) * PITCH + a3];
  P1[(size_t)0 * P1SZ + (size_t)nd * 4096 + hi * 256 + a2 * 16 + a3] = s0;
}

// P2 from P1 (6 tensors), P3 from P2 (4 tensors): cheap strided reductions.
__global__ void pool2(const float* __restrict__ P1, float* __restrict__ P2) {
  int idx = blockIdx.x * 256 + threadIdx.x;          // 6*1024*256
  int c = idx / P2SZ;
  int r = idx - c * P2SZ;
  int nd = r >> 8, o = r & 255;
  const int src[6] = {0, 0, 0, 1, 1, 2};             // source P1 for A={p,q}
  const int sd[6]  = {0, 1, 2, 1, 2, 2};             // dim of P1_p to reduce
  const float* base = P1 + (size_t)src[c] * P1SZ + (size_t)nd * 4096;
  int o0 = o >> 4, o1 = o & 15, stride, off;
  if (sd[c] == 0)      { stride = 256; off = o0 * 16 + o1; }
  else if (sd[c] == 1) { stride = 16;  off = o0 * 256 + o1; }
  else                 { stride = 1;   off = o0 * 256 + o1 * 16; }
  float s = 0.f;
#pragma unroll
  for (int i = 0; i < 16; ++i) s += base[off + i * stride];
  P2[(size_t)c * P2SZ + (size_t)nd * 256 + o] = s;
}

__global__ void pool3(const float* __restrict__ P2, float* __restrict__ P3) {
  int idx = blockIdx.x * 256 + threadIdx.x;          // 4*1024*16
  int c = idx / P3SZ;
  int r = idx - c * P3SZ;
  int nd = r >> 4, o = r & 15;
  const int src[4] = {0, 0, 1, 3};
  const int sd[4]  = {0, 1, 1, 1};
  const float* base = P2 + (size_t)src[c] * P2SZ + (size_t)nd * 256;
  int stride = (sd[c] == 0) ? 16 : 1;
  int off    = (sd[c] == 0) ? o : o * 16;
  float s = 0.f;
#pragma unroll
  for (int i = 0; i < 16; ++i) s += base[off + i * stride];
  P3[(size_t)c * P3SZ + nd * 16 + o] = s;
}

// ---------------------------------------------------------------------------
// WMMA helpers. Fragment layout per CDNA5 ISA 7.12.2:
//   A: lane L holds A[M=L&15][K = 2*(L>>4) + {0,1}]  (VGPR pair)
//   B: lane L holds B[K = 2*(L>>4) + {0,1}][N=L&15]
//   C: VGPR r holds M = r + 8*(L>>4), N = L&15
// A-panel (16 s-rows x K=128 of W[b][d][s]) cached in 64 VGPRs per wave.
__device__ __forceinline__ void load_apanel(const float* __restrict__ Wb, int s0,
                                            v2f* __restrict__ af) {
  int lane = threadIdx.x & 31;
  int col = lane & 15;
  int half = (lane >> 4) * 2;                        // 0 or 2
  const float* ap = Wb + half * 128 + s0 + col;      // A[M][K] = W[K][M]
#pragma unroll
  for (int k = 0; k < 32; ++k) { af[k].x = ap[0]; af[k].y = ap[128]; ap += 512; }
}

__device__ __forceinline__ v8f wmma_b(const v2f* __restrict__ af,
                                      const float* __restrict__ X, size_t P,
                                      int pos0, v8f c) {
  int lane = threadIdx.x & 31;
  int col = lane & 15;
  int half = (lane >> 4) * 2;
  const float* bp = X + (size_t)half * P + pos0 + col;
#pragma unroll
  for (int k = 0; k < 32; ++k) {
    v2f bv;
    bv.x = bp[0]; bv.y = bp[P];
    c = __builtin_amdgcn_wmma_f32_16x16x4_f32(false, af[k], false, bv,
                                              (short)0, c, false, false);
    bp += 4 * P;
  }
  return c;
}

// T1_t[n,s,i] = sum_{A3} W[53+a*4+t]^T P3_a ; 4 t, 8 n; 8 waves = s-blocks.
__global__ void gemm_t1(const float* __restrict__ Wp, const float* __restrict__ P3,
                        float* __restrict__ T1) {
  int blk = blockIdx.x;                              // 32 = 4t * 8n
  int t = blk >> 3, n = blk & 7;
  int wave = threadIdx.x >> 5, lane = threadIdx.x & 31;
  v2f af[32];
  v8f c = {};
  for (int a = 0; a < 4; ++a) {
    load_apanel(Wp + (size_t)(53 + a * 4 + t) * 16384, wave * 16, af);
    c = wmma_b(af, P3 + (size_t)a * P3SZ + (size_t)n * 128 * 16, 16, 0, c);
  }
  int col = lane & 15, rb = (lane >> 4) * 8;
#pragma unroll
  for (int r = 0; r < 8; ++r) {
    int s = wave * 16 + rb + r;
    T1[(size_t)t * P3SZ + ((size_t)n * 128 + s) * 16 + col] = c[r];
  }
}

// T2_t[n,s,o] = sum_{A2} W[17+a*6+t]^T P2_a ; 6 t, 8 n, 16 pos-tiles.
__global__ void gemm_t2(const float* __restrict__ Wp, const float* __restrict__ P2,
                        float* __restrict__ T2) {
  int blk = blockIdx.x;                              // 768
  int t = blk / 128, rem = blk - t * 128;
  int n = rem >> 4, pt = rem & 15;
  int wave = threadIdx.x >> 5, lane = threadIdx.x & 31;
  v2f af[32];
  v8f c = {};
  for (int a = 0; a < 6; ++a) {
    load_apanel(Wp + (size_t)(17 + a * 6 + t) * 16384, wave * 16, af);
    c = wmma_b(af, P2 + (size_t)a * P2SZ + (size_t)n * 128 * 256, 256, pt * 16, c);
  }
  int col = lane & 15, rb = (lane >> 4) * 8, pos = pt * 16 + col;
#pragma unroll
  for (int r = 0; r < 8; ++r) {
    int s = wave * 16 + rb + r;
    T2[(size_t)t * P2SZ + ((size_t)n * 128 + s) * 256 + pos] = c[r];
  }
}

// T3_t[n,s,p] = sum_{a} W[1+a*4+t]^T P1_a ; 2 pos-tiles per wave, sources
// outer so each A-panel load is amortized over both accumulators.
// Epilogue folds T2/T1/bias broadcasts.
__global__ void gemm_t3(const float* __restrict__ Wp, const float* __restrict__ P1,
                        const float* __restrict__ T2, const float* __restrict__ T1,
                        const float* __restrict__ bias, float* __restrict__ T3) {
  int blk = blockIdx.x;                              // 4096 = 4t * 8n * 128pg
  int t = blk >> 10, rem = blk & 1023;
  int n = rem >> 7, pg = rem & 127;
  int wave = threadIdx.x >> 5, lane = threadIdx.x & 31;
  int posb = pg * 32;
  v2f af[32];
  v8f c0 = {}, c1 = {};
  for (int a = 0; a < 4; ++a) {
    load_apanel(Wp + (size_t)(1 + a * 4 + t) * 16384, wave * 16, af);
    const float* X = P1 + (size_t)a * P1SZ + (size_t)n * 128 * 4096;
    c0 = wmma_b(af, X, 4096, posb, c0);
    c1 = wmma_b(af, X, 4096, posb + 16, c1);
  }
  int col = lane & 15, rb = (lane >> 4) * 8;
#pragma unroll
  for (int half = 0; half < 2; ++half) {
    int pos = posb + half * 16 + col;
    int c2 = pos & 15, c1i = (pos >> 4) & 15, c0i = pos >> 8;
    v8f& acc = half ? c1 : c0;
#pragma unroll
    for (int r = 0; r < 8; ++r) {
      int s = wave * 16 + rb + r;
      size_t ns = (size_t)n * 128 + s;
      float e = 0.f;
      if (t == 0) {                                  // B={0,1,2}
        e += T2[(size_t)0 * P2SZ + ns * 256 + c0i * 16 + c1i];
        e += T2[(size_t)1 * P2SZ + ns * 256 + c0i * 16 + c2];
        e += T2[(size_t)3 * P2SZ + ns * 256 + c1i * 16 + c2];
        e += T1[(size_t)0 * P3SZ + ns * 16 + c0i];
      } else if (t == 1) {                           // B={0,1,3}
        e += T2[(size_t)2 * P2SZ + ns * 256 + c0i * 16 + c2];
        e += T2[(size_t)4 * P2SZ + ns * 256 + c1i * 16 + c2];
        e += T1[(size_t)1 * P3SZ + ns * 16 + c1i];
      } else if (t == 2) {                           // B={0,2,3}
        e += T2[(size_t)5 * P2SZ + ns * 256 + c1i * 16 + c2];
        e += T1[(size_t)2 * P3SZ + ns * 16 + c1i];
      } else {                                       // B={1,2,3}
        e += T1[(size_t)3 * P3SZ + ns * 16 + c2];
        e += bias[s];
      }
      T3[(size_t)t * P1SZ + ns * 4096 + pos] = acc[r] + e;
    }
  }
}

// Main: out[n,s,p] = W[0]^T x[n] + 4 T3 broadcasts. One A-panel per wave
// amortized over 4 pos-tiles; 8 waves per block share B addresses (WGP$).
__global__ void gemm_main(const float* __restrict__ Wp, const float* __restrict__ x,
                          const float* __restrict__ T3, float* __restrict__ out) {
  int blk = blockIdx.x;                              // 8192 = 8n * 1024pg
  int n = blk >> 10, pg = blk & 1023;
  int wave = threadIdx.x >> 5, lane = threadIdx.x & 31;
  int col = lane & 15, rb = (lane >> 4) * 8;
  v2f af[32];
  load_apanel(Wp, wave * 16, af);                    // b = 0 panel, reused 4x
  const float* X = x + (size_t)n * 128 * 65536;
#pragma unroll
  for (int tt = 0; tt < 4; ++tt) {
    int pos0 = pg * 64 + tt * 16;
    v8f c = {};
    c = wmma_b(af, X, 65536, pos0, c);
    int pos = pos0 + col;
    int c3 = pos & 15, c2 = (pos >> 4) & 15, c1 = (pos >> 8) & 15, c0 = pos >> 12;
#pragma unroll
    for (int r = 0; r < 8; ++r) {
      int s = wave * 16 + rb + r;
      size_t ns = (size_t)n * 128 + s;
      const float* t3 = T3 + ns * 4096;
      float e = t3[(size_t)0 * P1SZ + c0 * 256 + c1 * 16 + c2]
              + t3[(size_t)1 * P1SZ + c0 * 256 + c1 * 16 + c3]
              + t3[(size_t)2 * P1SZ + c0 * 256 + c2 * 16 + c3]
              + t3[(size_t)3 * P1SZ + c1 * 256 + c2 * 16 + c3];
      out[ns * 65536 + pos] = c[r] + e;
    }
  }
}

// ---------------------------------------------------------------------------
extern "C" void kernel_launch(void* const* d_in, const int* in_sizes, int n_in,
                              void* d_out, int out_size, void* d_ws, size_t ws_size,
                              hipStream_t stream) {
  (void)in_sizes; (void)n_in; (void)out_size; (void)ws_size;
  const float* x     = (const float*)d_in[0];
  const float* coefs = (const float*)d_in[1];
  const float* bias  = (const float*)d_in[2];
  float* out = (float*)d_out;
  float* ws  = (float*)d_ws;

  float* Wp = ws;                      // 1,130,496 floats
  float* P1 = Wp + 1130496;            // 16,777,216
  float* P2 = P1 + 4 * (size_t)P1SZ;   // 1,572,864
  float* P3 = P2 + 6 * (size_t)P2SZ;   // 65,536
  float* T3 = P3 + 4 * (size_t)P3SZ;   // 16,777,216
  float* T2 = T3 + 4 * (size_t)P1SZ;   // 1,572,864
  float* T1 = T2 + 6 * (size_t)P2SZ;   // 65,536

  pack_w   <<<4416, 256, 0, stream>>>(coefs, Wp);
  pool_a   <<<16384, 256, 0, stream>>>(x, P1);
  pool_b   <<<16384, 256, 0, stream>>>(x, P1);
  pool2    <<<6144, 256, 0, stream>>>(P1, P2);
  pool3    <<<256, 256, 0, stream>>>(P2, P3);
  gemm_t1  <<<32, 256, 0, stream>>>(Wp, P3, T1);
  gemm_t2  <<<768, 256, 0, stream>>>(Wp, P2, T2);
  gemm_t3  <<<4096, 256, 0, stream>>>(Wp, P1, T2, T1, bias, T3);
  gemm_main<<<8192, 256, 0, stream>>>(Wp, x, T3, out);
}